// CostSensitiveLoss_7911329759639
// MI455X (gfx1250) — compile-verified
//
#include <hip/hip_runtime.h>
#include <hip/hip_bf16.h>

#define NUM_C 1024
#define ROWS_PER_BLOCK 8   // 8 wave32s per block, one row per wave

// ---------------------------------------------------------------------------
// Kernel 1: one wave32 per row.
//   - gfx1250 async DMA: global -> LDS (global_load_async_to_lds_b128),
//     8 x 512B per lane-group = 4 KB row, tracked by ASYNCcnt.
//   - register-resident row (8 x float4 per lane), wave shuffle reductions.
//   - per-block partial loss written to workspace.
// ---------------------------------------------------------------------------
__global__ __launch_bounds__(256) void cs_loss_rows(
    const float* __restrict__ x,      // [B, 1024]
    const int*   __restrict__ tgt,    // [B]
    float*       __restrict__ partial)// [gridDim.x]
{
    __shared__ float smem[ROWS_PER_BLOCK * NUM_C];   // 32 KB staging
    __shared__ float wave_loss[ROWS_PER_BLOCK];

    const int tid  = threadIdx.x;
    const int wave = tid >> 5;
    const int lane = tid & 31;
    const int row  = blockIdx.x * ROWS_PER_BLOCK + wave;

    // --- async DMA the row into LDS -------------------------------------
    // lane l covers float4 elements {l + 32k : k=0..7}; immediate offset
    // k*512 bytes applies to BOTH the global and LDS address (ISA 10.x).
    const float*   gbase   = x + (size_t)row * NUM_C + lane * 4;
    unsigned int   ldsaddr = (unsigned int)(size_t)(&smem[wave * NUM_C + lane * 4]);

    asm volatile(
        "global_load_async_to_lds_b128 %0, %1, off\n\t"
        "global_load_async_to_lds_b128 %0, %1, off offset:512\n\t"
        "global_load_async_to_lds_b128 %0, %1, off offset:1024\n\t"
        "global_load_async_to_lds_b128 %0, %1, off offset:1536\n\t"
        "global_load_async_to_lds_b128 %0, %1, off offset:2048\n\t"
        "global_load_async_to_lds_b128 %0, %1, off offset:2560\n\t"
        "global_load_async_to_lds_b128 %0, %1, off offset:3072\n\t"
        "global_load_async_to_lds_b128 %0, %1, off offset:3584\n\t"
        "s_wait_asynccnt 0"
        :
        : "v"(ldsaddr), "v"(gbase)
        : "memory");

    const int   t  = tgt[row];
    const float tf = (float)t;

    // --- pull row fragment into registers (ds_load_b128, conflict-free) --
    float4 v[8];
#pragma unroll
    for (int k = 0; k < 8; ++k) {
        v[k] = *reinterpret_cast<const float4*>(
                   &smem[wave * NUM_C + lane * 4 + k * 128]);
    }

    // --- row max (lane-local then wave32 xor-shuffle) ---------------------
    float m = -3.402823466e38f;
#pragma unroll
    for (int k = 0; k < 8; ++k) {
        m = fmaxf(m, fmaxf(fmaxf(v[k].x, v[k].y), fmaxf(v[k].z, v[k].w)));
    }
#pragma unroll
    for (int off = 16; off > 0; off >>= 1)
        m = fmaxf(m, __shfl_xor(m, off, 32));

    // --- sum exp(x-m) and weighted sum w*x, w = |t-j|+1 -------------------
    float s  = 0.0f;
    float ws = 0.0f;
#pragma unroll
    for (int k = 0; k < 8; ++k) {
        const float jb = (float)(lane * 4 + k * 128);
        float xv, w;

        xv = v[k].x; w = fabsf(tf - (jb + 0.0f)) + 1.0f;
        s += __expf(xv - m); ws = fmaf(w, xv, ws);
        xv = v[k].y; w = fabsf(tf - (jb + 1.0f)) + 1.0f;
        s += __expf(xv - m); ws = fmaf(w, xv, ws);
        xv = v[k].z; w = fabsf(tf - (jb + 2.0f)) + 1.0f;
        s += __expf(xv - m); ws = fmaf(w, xv, ws);
        xv = v[k].w; w = fabsf(tf - (jb + 3.0f)) + 1.0f;
        s += __expf(xv - m); ws = fmaf(w, xv, ws);
    }
#pragma unroll
    for (int off = 16; off > 0; off >>= 1) {
        s  += __shfl_xor(s,  off, 32);
        ws += __shfl_xor(ws, off, 32);
    }

    if (lane == 0) {
        // W = sum_j (|t-j|+1) = C + t(t+1)/2 + (C-1-t)(C-t)/2  (exact in f32)
        const float Wsum = (float)NUM_C
                         + 0.5f * tf * (tf + 1.0f)
                         + 0.5f * (float)(NUM_C - 1 - t) * (float)(NUM_C - t);
        const float lse  = m + __logf(s);
        wave_loss[wave]  = lse * Wsum - ws;   // = -sum_j w_j * logp_j
    }
    __syncthreads();

    if (tid == 0) {
        float acc = 0.0f;
#pragma unroll
        for (int i = 0; i < ROWS_PER_BLOCK; ++i) acc += wave_loss[i];
        partial[blockIdx.x] = acc;
    }
}

// ---------------------------------------------------------------------------
// Kernel 2: deterministic single-block reduction of partials -> mean.
// Double accumulation (32768 losses of magnitude ~1e6 each).
// ---------------------------------------------------------------------------
__global__ __launch_bounds__(1024) void cs_loss_reduce(
    const float* __restrict__ partial, float* __restrict__ out,
    int n, double invB)
{
    __shared__ double red[1024];
    double acc = 0.0;
    for (int i = threadIdx.x; i < n; i += 1024)
        acc += (double)partial[i];
    red[threadIdx.x] = acc;
    __syncthreads();
#pragma unroll
    for (int s = 512; s > 0; s >>= 1) {
        if ((int)threadIdx.x < s) red[threadIdx.x] += red[threadIdx.x + s];
        __syncthreads();
    }
    if (threadIdx.x == 0)
        out[0] = (float)(red[0] * invB);
}

// ---------------------------------------------------------------------------
extern "C" void kernel_launch(void* const* d_in, const int* in_sizes, int n_in,
                              void* d_out, int out_size, void* d_ws, size_t ws_size,
                              hipStream_t stream) {
    const float* x   = (const float*)d_in[0];   // [B, 1024] f32 logits
    const int*   tgt = (const int*)d_in[1];     // [B] labels
    float*       ws  = (float*)d_ws;
    float*       out = (float*)d_out;

    const int B       = in_sizes[0] / NUM_C;    // 32768
    const int nblocks = B / ROWS_PER_BLOCK;     // 4096

    cs_loss_rows<<<nblocks, 256, 0, stream>>>(x, tgt, ws);
    cs_loss_reduce<<<1, 1024, 0, stream>>>(ws, out, nblocks, 1.0 / (double)B);
}